// KDEECE_35897336660463
// MI455X (gfx1250) — compile-verified
//
#include <hip/hip_runtime.h>
#include <math.h>

typedef __attribute__((ext_vector_type(2))) float v2f;
typedef __attribute__((ext_vector_type(8))) float v8f;

#define N_ROWS 8192
#define N_CLS  10
#define CPAD   12              // K padded to 12 (3 x K=4 WMMA steps)
#define NTILE  (N_ROWS / 16)   // 512 tiles of 16 rows
#define WPB    4               // waves per block (one i-tile per wave)

// ---------------------------------------------------------------------------
// Phase 1a: softmax, log-probs, log-beta per row. logβ_j = Σ lgamma(10f+1) - lgamma(20)
// ---------------------------------------------------------------------------
__global__ __launch_bounds__(256) void prep_kernel(const float* __restrict__ logits,
                                                   float* __restrict__ fpad,
                                                   float* __restrict__ lfpad,
                                                   float* __restrict__ logbeta)
{
    int r = blockIdx.x * blockDim.x + threadIdx.x;
    if (r >= N_ROWS) return;
    float l[N_CLS];
    float m = -3.4e38f;
    #pragma unroll
    for (int c = 0; c < N_CLS; ++c) { l[c] = logits[r * N_CLS + c]; m = fmaxf(m, l[c]); }
    float s = 0.f;
    #pragma unroll
    for (int c = 0; c < N_CLS; ++c) { l[c] = expf(l[c] - m); s += l[c]; }
    float inv = 1.0f / s;
    float lb = 0.f;
    #pragma unroll
    for (int c = 0; c < N_CLS; ++c) {
        float f = l[c] * inv;
        fpad [r * CPAD + c] = f;
        lfpad[r * CPAD + c] = logf(f + 1e-45f);
        lb += lgammaf(fmaf(10.f, f, 1.f));
    }
    fpad [r * CPAD + 10] = 0.f;  fpad [r * CPAD + 11] = 0.f;
    lfpad[r * CPAD + 10] = 0.f;  lfpad[r * CPAD + 11] = 0.f;
    logbeta[r] = lb - lgammaf(20.0f);
}

// ---------------------------------------------------------------------------
// Phase 1b: precompute augmented one-hot B-fragments Y' for GEMM2.
//   Y'[j][c] = (c < 10 ? labels[j]==c : c==10)   (col 10 = ones -> row sum/den)
// ---------------------------------------------------------------------------
__global__ __launch_bounds__(256) void yfrag_kernel(const int* __restrict__ labels,
                                                    v2f* __restrict__ yfrag)
{
    int idx = blockIdx.x * blockDim.x + threadIdx.x;   // 0 .. NTILE*4*32-1
    if (idx >= NTILE * 4 * 32) return;
    int lane = idx & 31;
    int kk   = (idx >> 5) & 3;
    int jt   = idx >> 7;
    int n    = lane & 15;
    int h    = lane >> 4;
    int j1   = jt * 16 + kk * 4 + 2 * h;
    int l0   = labels[j1];
    int l1   = labels[j1 + 1];
    v2f b;
    b.x = (n == 10) ? 1.f : ((l0 == n) ? 1.f : 0.f);
    b.y = (n == 10) ? 1.f : ((l1 == n) ? 1.f : 0.f);
    yfrag[idx] = b;
}

// ---------------------------------------------------------------------------
// One 16x16 (i-tile x j-tile) step:
//   GEMM1 (K=12): dot = logf_i . f_j;  kern = exp(10*dot - logbeta_j)
//   DIAG: zero k(i,i) (only instantiated for the jt == it tile)
//   GEMM2 (K=16): acc2 += kern_tile x Y'
// ---------------------------------------------------------------------------
template <bool DIAG>
__device__ __forceinline__ void process_tile(int jt, int n, int h, int lane,
                                             const float* __restrict__ fpad,
                                             const float* __restrict__ logbeta,
                                             const v2f*   __restrict__ yfrag,
                                             const v2f a0[3],
                                             float (* __restrict__ kt)[18],
                                             v8f& acc2)
{
    const int j0 = jt * 16;

    v8f acc = {};
    #pragma unroll
    for (int kk = 0; kk < 3; ++kk) {
        v2f b = *(const v2f*)&fpad[(j0 + n) * CPAD + kk * 4 + 2 * h];
        acc = __builtin_amdgcn_wmma_f32_16x16x4_f32(false, a0[kk], false, b,
                                                    (short)0, acc, false, false);
    }

    const float lb = logbeta[j0 + n];
    #pragma unroll
    for (int v = 0; v < 8; ++v) {
        float k = __expf(fmaf(10.f, acc[v], -lb));   // native v_exp_f32
        if (DIAG) {
            const int m = v + 8 * h;                 // i0 == j0 here: diag at m == n
            k = (m == n) ? 0.f : k;
        }
        kt[v + 8 * h][n] = k;
    }
    asm volatile("s_wait_dscnt 0" ::: "memory");     // same-wave DS RAW fence

    #pragma unroll
    for (int kk = 0; kk < 4; ++kk) {
        v2f a2;
        a2.x = kt[n][kk * 4 + 2 * h];
        a2.y = kt[n][kk * 4 + 2 * h + 1];
        v2f b2 = yfrag[(jt * 4 + kk) * 32 + lane];
        acc2 = __builtin_amdgcn_wmma_f32_16x16x4_f32(false, a2, false, b2,
                                                     (short)0, acc2, false, false);
    }
}

// ---------------------------------------------------------------------------
// Phase 2: per i-tile, sweep all j-tiles (j-loop split around the diagonal tile)
// ---------------------------------------------------------------------------
__global__ __launch_bounds__(32 * WPB) void kde_main(const float* __restrict__ fpad,
                                                     const float* __restrict__ lfpad,
                                                     const float* __restrict__ logbeta,
                                                     const v2f*   __restrict__ yfrag,
                                                     float*       __restrict__ partials)
{
    __shared__ float ktile[WPB][16][18];   // stride 18: conflict-free rows & cols

    const int lane = threadIdx.x & 31;
    // wave index is wave-uniform: pin it to an SGPR so the split j-loops get
    // scalar (s_cbranch) control instead of per-lane EXEC-mask loop control.
    const int wave = __builtin_amdgcn_readfirstlane((int)(threadIdx.x >> 5));
    const int it   = blockIdx.x * WPB + wave;   // i-tile id (SGPR), 0..511
    const int i0   = it * 16;
    const int n    = lane & 15;                 // A-row / B-col / D-col index
    const int h    = lane >> 4;                 // lane half

    // A fragments for GEMM1: rows of log f for this i-tile (constant over j loop)
    v2f a0[3];
    #pragma unroll
    for (int kk = 0; kk < 3; ++kk)
        a0[kk] = *(const v2f*)&lfpad[(i0 + n) * CPAD + kk * 4 + 2 * h];

    float (*kt)[18] = ktile[wave];
    v8f acc2 = {};   // D layout: row i0 + v + 8h, col c = n  (c==10 -> den)

    #pragma unroll 2
    for (int jt = 0; jt < it; ++jt)
        process_tile<false>(jt, n, h, lane, fpad, logbeta, yfrag, a0, kt, acc2);

    process_tile<true>(it, n, h, lane, fpad, logbeta, yfrag, a0, kt, acc2);

    #pragma unroll 2
    for (int jt = it + 1; jt < NTILE; ++jt)
        process_tile<false>(jt, n, h, lane, fpad, logbeta, yfrag, a0, kt, acc2);

    // ---- finalize: per_sample = Σ_c | kern_y/den - f |, reduce over tile ----
    float t = 0.f;
    #pragma unroll
    for (int v = 0; v < 8; ++v) {
        float den = __shfl(acc2[v], (lane & 16) | 10, 32);  // col 10 of own row
        den = fmaxf(den, 1e-10f);
        float ratio = acc2[v] / den;
        float contrib = 0.f;
        if (n < N_CLS) {
            const int irow = i0 + v + 8 * h;
            contrib = fabsf(ratio - fpad[irow * CPAD + n]);
        }
        t += contrib;
    }
    #pragma unroll
    for (int off = 16; off > 0; off >>= 1) t += __shfl_xor(t, off, 32);
    if (lane == 0) partials[it] = t;
}

// ---------------------------------------------------------------------------
// Phase 3: deterministic tree reduction of 512 tile partials -> mean
// ---------------------------------------------------------------------------
__global__ __launch_bounds__(256) void reduce_kernel(const float* __restrict__ partials,
                                                     float* __restrict__ out)
{
    __shared__ float s[256];
    int t = threadIdx.x;
    s[t] = partials[t] + partials[t + 256];
    __syncthreads();
    #pragma unroll
    for (int off = 128; off > 0; off >>= 1) {
        if (t < off) s[t] += s[t + off];
        __syncthreads();
    }
    if (t == 0) out[0] = s[0] * (1.0f / (float)N_ROWS);
}

// ---------------------------------------------------------------------------
extern "C" void kernel_launch(void* const* d_in, const int* in_sizes, int n_in,
                              void* d_out, int out_size, void* d_ws, size_t ws_size,
                              hipStream_t stream)
{
    const float* logits = (const float*)d_in[0];
    const int*   labels = (const int*)d_in[1];

    float* fpad     = (float*)d_ws;                 // N * 12
    float* lfpad    = fpad    + N_ROWS * CPAD;      // N * 12
    float* logbeta  = lfpad   + N_ROWS * CPAD;      // N
    float* partials = logbeta + N_ROWS;             // NTILE
    v2f*   yfrag    = (v2f*)(partials + NTILE);     // NTILE*4*32 float2 (512 KB)

    prep_kernel  <<<N_ROWS / 256, 256, 0, stream>>>(logits, fpad, lfpad, logbeta);
    yfrag_kernel <<<(NTILE * 4 * 32) / 256, 256, 0, stream>>>(labels, yfrag);
    kde_main     <<<NTILE / WPB, 32 * WPB, 0, stream>>>(fpad, lfpad, logbeta, yfrag, partials);
    reduce_kernel<<<1, 256, 0, stream>>>(partials, (float*)d_out);
}